// LabelAttention_30081950941746
// MI455X (gfx1250) — compile-verified
//
#include <hip/hip_runtime.h>
#include <hip/hip_bf16.h>
#include <math.h>

// ---------------------------------------------------------------------------
// LabelAttention, algebraically collapsed (see analysis):
//   A[l]      = concat(w_ks1[l] @ w_qs1[l], w_ks2[l] @ w_qs2[l])   (112 x 1024)
//   scores    = (inp @ A^T) / 32                                   (112 x 1024)
//   attn      = softmax over seq=128 within each (l, b)
//   cbar[l,b] = attn[l,b,:] @ inp_block[b]                         (112 x 8 x 1024)
//   ctx1/ctx2 = cbar halves @ w_vs1[l] / w_vs2[l]                  (8 x 64 per l)
//   P1        = proj1_w @ reduce_w[:512], P2 = proj2_w @ reduce_w[512:]
//   comb_red  = ctx1 @ P1 + ctx2 @ P2                              (112 x 8 x 64)
//   inp_red   = inp @ reduce_w                                     (1024 x 64)
//   red[l,t]  = comb_red[l, t/128] + inp_red[t];  out = LN(red) interleaved
// Memory-bound (~95 MB total traffic); f32 precision kept throughout, GEMMs
// run on V_WMMA_F32_16X16X4_F32.
// ---------------------------------------------------------------------------

#define D_MODEL   1024
#define D_CONTENT 512
#define D_POS     512
#define DK2       64
#define DV2       64
#define D_L       112
#define D_PROJ    64
#define MB        8
#define SEQ       128
#define T_TOK     1024
#define EPS_LN    1e-3f

typedef __attribute__((ext_vector_type(2))) float v2f;
typedef __attribute__((ext_vector_type(8))) float v8f;

// D(16x16,f32) = A(16x4,f32) * B(4x16,f32) + C.
// 8-arg pattern as probe-confirmed for 16x16xN shapes:
// (neg_a, A, neg_b, B, c_mod, C, reuse_a, reuse_b)
__device__ __forceinline__ v8f wmma_f32(v2f a, v2f b, v8f c) {
    return __builtin_amdgcn_wmma_f32_16x16x4_f32(false, a, false, b, (short)0, c,
                                                 false, false);
}

// --------------------------------------------------------------------------
// K1: A_mat[l][i] = sum_k w_ks{1,2}[l][i][k] * w_qs{1,2}[l][k]
// grid (112, 4) x 256 threads; reads w_ks1/w_ks2 exactly once (29 MB).
// --------------------------------------------------------------------------
__global__ void k_label_vec(const float* __restrict__ w_qs1, const float* __restrict__ w_qs2,
                            const float* __restrict__ w_ks1, const float* __restrict__ w_ks2,
                            float* __restrict__ A_mat) {
    int l = blockIdx.x;
    int i = blockIdx.y * blockDim.x + threadIdx.x;   // 0..1023
    const float* q;
    const float* w;
    if (i < D_CONTENT) {
        q = w_qs1 + l * DK2;
        w = w_ks1 + ((size_t)l * D_CONTENT + i) * DK2;
    } else {
        q = w_qs2 + l * DK2;
        w = w_ks2 + ((size_t)l * D_POS + (i - D_CONTENT)) * DK2;
    }
    float s = 0.f;
#pragma unroll
    for (int k = 0; k < DK2; ++k) s += w[k] * q[k];
    A_mat[(size_t)l * D_MODEL + i] = s;
}

// --------------------------------------------------------------------------
// K2: P1 = proj1_w @ reduce_w[:512,:], P2 = proj2_w @ reduce_w[512:,:]
// grid (16, 2) x 256 threads.
// --------------------------------------------------------------------------
__global__ void k_projred(const float* __restrict__ proj1_w, const float* __restrict__ proj2_w,
                          const float* __restrict__ reduce_w,
                          float* __restrict__ P1, float* __restrict__ P2) {
    int h   = blockIdx.y;                                // 0 -> P1, 1 -> P2
    int idx = blockIdx.x * blockDim.x + threadIdx.x;     // 0..4095
    int i = idx >> 6, j = idx & 63;
    const float* pw = h ? proj2_w : proj1_w;
    const float* rw = reduce_w + (h ? (size_t)D_CONTENT * D_PROJ : 0);
    float s = 0.f;
    for (int c = 0; c < D_CONTENT; ++c) s += pw[i * D_CONTENT + c] * rw[(size_t)c * D_PROJ + j];
    (h ? P2 : P1)[i * D_PROJ + j] = s;
}

// --------------------------------------------------------------------------
// K3 (WMMA): inp_red = inp(1024x1024) @ reduce_w(1024x64).
// 8 waves/block, one 16x16 C-tile per wave. grid 32 x 256.
// --------------------------------------------------------------------------
__global__ void k_inp_red_wmma(const float* __restrict__ inp, const float* __restrict__ reduce_w,
                               float* __restrict__ inp_red) {
    int lane = threadIdx.x & 31, wave = threadIdx.x >> 5;
    int tile = blockIdx.x * 8 + wave;           // 0..255 (64 M-tiles x 4 N-tiles)
    int mtile = tile >> 2, ntile = tile & 3;
    int mn = lane & 15, khalf = lane >> 4;      // A uses m=mn, B uses n=mn
    v8f acc = {0.f, 0.f, 0.f, 0.f, 0.f, 0.f, 0.f, 0.f};
    const float* arow = inp + (size_t)(mtile * 16 + mn) * D_MODEL;
    for (int k = 0; k < D_MODEL; k += 4) {
        int kk = k + khalf * 2;
        v2f a; a.x = arow[kk];                          a.y = arow[kk + 1];
        v2f b; b.x = reduce_w[(size_t)kk * D_PROJ + ntile * 16 + mn];
               b.y = reduce_w[(size_t)(kk + 1) * D_PROJ + ntile * 16 + mn];
        acc = wmma_f32(a, b, acc);
    }
#pragma unroll
    for (int r = 0; r < 8; ++r) {
        int row = mtile * 16 + r + khalf * 8;
        inp_red[(size_t)row * D_PROJ + ntile * 16 + mn] = acc[r];
    }
}

// --------------------------------------------------------------------------
// K4 (WMMA): scores = (A_mat(112x1024) @ inp^T(1024x1024)) / 32.
// 7 x 64 tiles -> grid 56 x 256.
// --------------------------------------------------------------------------
__global__ void k_scores_wmma(const float* __restrict__ A_mat, const float* __restrict__ inp,
                              float* __restrict__ scores) {
    int lane = threadIdx.x & 31, wave = threadIdx.x >> 5;
    int tile = blockIdx.x * 8 + wave;            // 0..447
    int ltile = tile >> 6, ttile = tile & 63;
    int mn = lane & 15, khalf = lane >> 4;
    v8f acc = {0.f, 0.f, 0.f, 0.f, 0.f, 0.f, 0.f, 0.f};
    const float* arow = A_mat + (size_t)(ltile * 16 + mn) * D_MODEL;
    const float* brow = inp   + (size_t)(ttile * 16 + mn) * D_MODEL;  // B[k][n] = inp[n][k]
    for (int k = 0; k < D_MODEL; k += 4) {
        int kk = k + khalf * 2;
        v2f a; a.x = arow[kk]; a.y = arow[kk + 1];
        v2f b; b.x = brow[kk]; b.y = brow[kk + 1];
        acc = wmma_f32(a, b, acc);
    }
    const float sc = 0.03125f;                   // 1/sqrt(D_MODEL)
#pragma unroll
    for (int r = 0; r < 8; ++r) {
        int l = ltile * 16 + r + khalf * 8;
        scores[(size_t)l * T_TOK + ttile * 16 + mn] = acc[r] * sc;
    }
}

// --------------------------------------------------------------------------
// K5: softmax over seq=128 within each (l, b); in-place on scores.
// grid 896 x 128.
// --------------------------------------------------------------------------
__global__ void k_softmax(float* __restrict__ scores) {
    __shared__ float sd[SEQ];
    int l = blockIdx.x >> 3, b = blockIdx.x & 7;
    int t = threadIdx.x;
    size_t idx = (size_t)l * T_TOK + b * SEQ + t;
    float x = scores[idx];
    sd[t] = x; __syncthreads();
    for (int st = 64; st > 0; st >>= 1) { if (t < st) sd[t] = fmaxf(sd[t], sd[t + st]); __syncthreads(); }
    float mx = sd[0]; __syncthreads();
    float e = expf(x - mx);
    sd[t] = e; __syncthreads();
    for (int st = 64; st > 0; st >>= 1) { if (t < st) sd[t] += sd[t + st]; __syncthreads(); }
    scores[idx] = e / sd[0];
}

// --------------------------------------------------------------------------
// K6 (WMMA): cbar[l,b,:] = attn[l,b,:] @ inp_block[b]  (per b: 112x128 @ 128x1024)
// 8 b * 7 ltiles * 64 dtiles = 3584 tiles -> grid 448 x 256.
// --------------------------------------------------------------------------
__global__ void k_cbar_wmma(const float* __restrict__ attn, const float* __restrict__ inp,
                            float* __restrict__ cbar) {
    int lane = threadIdx.x & 31, wave = threadIdx.x >> 5;
    int tile = blockIdx.x * 8 + wave;
    int b = tile / 448;
    int rem = tile % 448;
    int ltile = rem >> 6, dtile = rem & 63;
    int mn = lane & 15, khalf = lane >> 4;
    v8f acc = {0.f, 0.f, 0.f, 0.f, 0.f, 0.f, 0.f, 0.f};
    const float* arow  = attn + (size_t)(ltile * 16 + mn) * T_TOK + b * SEQ;
    const float* bbase = inp  + (size_t)b * SEQ * D_MODEL + dtile * 16 + mn;
    for (int k = 0; k < SEQ; k += 4) {
        int kk = k + khalf * 2;
        v2f a;  a.x  = arow[kk];                      a.y  = arow[kk + 1];
        v2f bb; bb.x = bbase[(size_t)kk * D_MODEL];   bb.y = bbase[(size_t)(kk + 1) * D_MODEL];
        acc = wmma_f32(a, bb, acc);
    }
#pragma unroll
    for (int r = 0; r < 8; ++r) {
        int l = ltile * 16 + r + khalf * 8;
        cbar[((size_t)l * MB + b) * D_MODEL + dtile * 16 + mn] = acc[r];
    }
}

// --------------------------------------------------------------------------
// K7: per-label ctx = cbar halves @ w_vs{1,2}[l]; comb_red = ctx1@P1 + ctx2@P2.
// One block per label; reads w_vs exactly once (29 MB). grid 112 x 256.
// --------------------------------------------------------------------------
__global__ void k_ctx_comb(const float* __restrict__ cbar, const float* __restrict__ w_vs1,
                           const float* __restrict__ w_vs2, const float* __restrict__ P1,
                           const float* __restrict__ P2, float* __restrict__ comb_red) {
    __shared__ float ctx[2][MB][DV2];
    int l = blockIdx.x;
    for (int task = threadIdx.x; task < 2 * MB * DV2; task += blockDim.x) {
        int half = task >> 9, rem = task & 511, b = rem >> 6, j = rem & 63;
        const float* crow = cbar + ((size_t)l * MB + b) * D_MODEL + half * D_CONTENT;
        const float* w    = (half ? w_vs2 : w_vs1) + (size_t)l * D_CONTENT * DV2 + j;
        float s = 0.f;
        for (int c = 0; c < D_CONTENT; ++c) s += crow[c] * w[(size_t)c * DV2];
        ctx[half][b][j] = s;
    }
    __syncthreads();
    for (int o = threadIdx.x; o < MB * D_PROJ; o += blockDim.x) {
        int b = o >> 6, j = o & 63;
        float s = 0.f;
#pragma unroll 8
        for (int i = 0; i < DV2; ++i)
            s += ctx[0][b][i] * P1[i * D_PROJ + j] + ctx[1][b][i] * P2[i * D_PROJ + j];
        comb_red[((size_t)l * MB + b) * D_PROJ + j] = s;
    }
}

// --------------------------------------------------------------------------
// K8: red = comb_red(broadcast) + inp_red; LayerNorm (n-1 var, sqrt(var)+eps);
// out[t, l*64+j]. grid (1024, 112) x 64 threads.
// --------------------------------------------------------------------------
__global__ void k_ln_out(const float* __restrict__ comb_red, const float* __restrict__ inp_red,
                         const float* __restrict__ ln_a, const float* __restrict__ ln_b,
                         float* __restrict__ out) {
    __shared__ float s[D_PROJ];
    int t = blockIdx.x, l = blockIdx.y, j = threadIdx.x;
    int b = t >> 7;
    float x = comb_red[((size_t)l * MB + b) * D_PROJ + j] + inp_red[(size_t)t * D_PROJ + j];
    s[j] = x; __syncthreads();
    for (int st = 32; st > 0; st >>= 1) { if (j < st) s[j] += s[j + st]; __syncthreads(); }
    float mu = s[0] * (1.f / D_PROJ); __syncthreads();
    float d = x - mu;
    s[j] = d * d; __syncthreads();
    for (int st = 32; st > 0; st >>= 1) { if (j < st) s[j] += s[j + st]; __syncthreads(); }
    float var = s[0] * (1.f / (D_PROJ - 1));
    out[(size_t)t * (D_L * D_PROJ) + l * D_PROJ + j] = d / (sqrtf(var) + EPS_LN) * ln_a[j] + ln_b[j];
}

extern "C" void kernel_launch(void* const* d_in, const int* in_sizes, int n_in,
                              void* d_out, int out_size, void* d_ws, size_t ws_size,
                              hipStream_t stream) {
    const float* inp      = (const float*)d_in[0];
    const float* w_qs1    = (const float*)d_in[1];
    const float* w_qs2    = (const float*)d_in[2];
    const float* w_ks1    = (const float*)d_in[3];
    const float* w_ks2    = (const float*)d_in[4];
    const float* w_vs1    = (const float*)d_in[5];
    const float* w_vs2    = (const float*)d_in[6];
    const float* proj1_w  = (const float*)d_in[7];
    const float* proj2_w  = (const float*)d_in[8];
    const float* reduce_w = (const float*)d_in[9];
    const float* ln_a     = (const float*)d_in[10];
    const float* ln_b     = (const float*)d_in[11];
    float* out = (float*)d_out;

    // Workspace carve-up (floats); total 1,277,952 floats ~= 5.1 MB.
    float* ws       = (float*)d_ws;
    float* A_mat    = ws;                    // 112*1024 = 114688
    float* scores   = A_mat   + 114688;      // 112*1024 = 114688 (becomes attn in-place)
    float* inp_red  = scores  + 114688;      // 1024*64  = 65536
    float* P1       = inp_red + 65536;       // 64*64    = 4096
    float* P2       = P1      + 4096;        // 64*64    = 4096
    float* cbar     = P2      + 4096;        // 112*8*1024 = 917504
    float* comb_red = cbar    + 917504;      // 112*8*64   = 57344

    k_label_vec   <<<dim3(D_L, 4),   256, 0, stream>>>(w_qs1, w_qs2, w_ks1, w_ks2, A_mat);
    k_projred     <<<dim3(16, 2),    256, 0, stream>>>(proj1_w, proj2_w, reduce_w, P1, P2);
    k_inp_red_wmma<<<32,             256, 0, stream>>>(inp, reduce_w, inp_red);
    k_scores_wmma <<<56,             256, 0, stream>>>(A_mat, inp, scores);
    k_softmax     <<<896,            128, 0, stream>>>(scores);
    k_cbar_wmma   <<<448,            256, 0, stream>>>(scores, inp, cbar);
    k_ctx_comb    <<<D_L,            256, 0, stream>>>(cbar, w_vs1, w_vs2, P1, P2, comb_red);
    k_ln_out      <<<dim3(1024, D_L), 64, 0, stream>>>(comb_red, inp_red, ln_a, ln_b, out);
}